// CESLayer_90812788506716
// MI455X (gfx1250) — compile-verified
//
#include <hip/hip_runtime.h>
#include <stdint.h>

typedef float v2f __attribute__((ext_vector_type(2)));
typedef float v8f __attribute__((ext_vector_type(8)));

#define I_DIM 128
#define O_DIM 128

// ---------------------------------------------------------------------------
// Prep: compute log|w| and angle(w) from (w_real, w_imag) [O=128, I=128] and
// scatter them into WMMA B-fragment order so the main kernel's inner loop is
// pure ds_load_b64 + v_wmma.
//
// B-fragment layout for V_WMMA_F32_16X16X4_F32 (B is K=4 x N=16, 2 VGPRs):
//   lane = (k>=2 ? 16 : 0) + n ,  vgpr v = k & 1   (k = (lane>>4)*2 + v)
// Fragment storage (floats): ws[((kb*8 + nb)*32 + lane)*2 + v]
//   kb = i/4 (k-block), nb = o/16 (n-block); second matrix at +16384 floats.
// ---------------------------------------------------------------------------
__global__ void ces_prep_weights(const float* __restrict__ w_real,
                                 const float* __restrict__ w_imag,
                                 float* __restrict__ ws) {
  int idx = blockIdx.x * blockDim.x + threadIdx.x;
  if (idx >= O_DIM * I_DIM) return;
  int o = idx >> 7;          // output-channel (N dimension)
  int i = idx & (I_DIM - 1); // input-channel  (K dimension)
  float wr = w_real[idx];
  float wi = w_imag[idx];
  float la = 0.5f * logf(wr * wr + wi * wi); // log|w|
  float an = atan2f(wi, wr);                 // angle(w)
  int kb = i >> 2;
  int kc = i & 3;
  int lane = ((kc >> 1) << 4) + (o & 15);
  int v    = kc & 1;
  int nb   = o >> 4;
  int fidx = ((kb * 8 + nb) * 32 + lane) * 2 + v;
  ws[fidx]                 = la;
  ws[O_DIM * I_DIM + fidx] = an;
}

// ---------------------------------------------------------------------------
// Main: fused dual-GEMM + pointwise epilogue.
// Block = 256 threads = 8 wave32s; block covers 64 rows of x.
// Wave w: rows [ (w>>1)*16 , +16 ), output columns [ (w&1)*64 , +64 ).
// Weight frags staged global->LDS with CDNA5 async-to-LDS copies; A tile
// preloaded into registers; hot loop = paired ds_load_2addr_b64 + v_wmma.
// ---------------------------------------------------------------------------
__global__ void __launch_bounds__(256)
ces_main(const float* __restrict__ x,
         const float* __restrict__ wsfrag,
         const float* __restrict__ bias,
         float* __restrict__ out) {
  extern __shared__ float lds[]; // 32768 floats = 128 KB (both weight frags)

  const int tid = threadIdx.x;

  // ---- Stage both pre-swizzled weight matrices into LDS via async copies.
  // INST_OFFSET is applied to both the LDS and global addresses (ISA 08,
  // GLOBAL_LOAD_ASYNC_TO_LDS pseudocode), so one base VGPR pair + immediate
  // per 4 KB slice moves the whole 128 KB. Tracked with ASYNCcnt.
  {
    unsigned ldsAddr = (unsigned)(uintptr_t)(void*)lds + (unsigned)tid * 16u;
    unsigned gOff    = (unsigned)tid * 16u;
#pragma unroll
    for (int it = 0; it < 32; ++it) {
      asm volatile("global_load_async_to_lds_b128 %0, %1, %2 offset:%3"
                   :
                   : "v"(ldsAddr), "v"(gOff), "s"(wsfrag), "n"(it * 4096)
                   : "memory");
    }
    asm volatile("s_wait_asynccnt 0x0" ::: "memory");
  }
  __syncthreads();

  const int wave  = tid >> 5;
  const int lane  = tid & 31;
  const int mbase = (wave >> 1) << 4; // 0,16,32,48
  const int ohalf = wave & 1;
  const int rowBase = blockIdx.x * 64;

  // ---- Preload the whole 16x128 A tile in WMMA fragment form (64 VGPRs):
  // m = lane&15 selects row, (lane>>4)*2 + v selects k.
  const float* xrow = x + (size_t)(rowBase + mbase + (lane & 15)) * I_DIM
                        + ((lane >> 4) << 1);
  v2f afrag[32];
#pragma unroll
  for (int kb = 0; kb < 32; ++kb)
    afrag[kb] = *(const v2f*)(xrow + kb * 4);

  // Per-wave fragment bases. The total fragment pool is 128 KB, i.e. larger
  // than one 16-bit DS-offset window, so force TWO independent base address
  // registers (64 KB window each): hide offY's relation to offM from the
  // optimizer so it cannot re-merge them into base+big-VGPR-add sequences.
  unsigned offM = (unsigned)(((ohalf * 4) * 32 + lane) * 2); // float index
  unsigned offY = offM + (unsigned)(O_DIM * I_DIM);
  asm volatile("" : "+v"(offY)); // opaque: keep as its own DS base register
  const float* baseM = lds + offM; // log|w| frags, offsets 0..64256 bytes
  const float* baseY = lds + offY; // angle frags,  offsets 0..64256 bytes

  v8f accm[4] = {}; // magnitude-GEMM accumulators
  v8f accy[4] = {}; // angle-GEMM accumulators

#pragma unroll
  for (int kb = 0; kb < 32; ++kb) {
    v2f bm[4], by[4];
#pragma unroll
    for (int t = 0; t < 4; ++t) {
      bm[t] = *(const v2f*)(baseM + (kb * 8 + t) * 64);
      by[t] = *(const v2f*)(baseY + (kb * 8 + t) * 64);
    }
#pragma unroll
    for (int t = 0; t < 4; ++t) {
      accm[t] = __builtin_amdgcn_wmma_f32_16x16x4_f32(
          false, afrag[kb], false, bm[t], (short)0, accm[t], false, false);
      accy[t] = __builtin_amdgcn_wmma_f32_16x16x4_f32(
          false, afrag[kb], false, by[t], (short)0, accy[t], false, false);
    }
  }

  // ---- Epilogue: C/D layout -> M = r + (lane>=16 ? 8 : 0), N = lane&15.
  // Hardware transcendentals (V_COS_F32 / V_EXP_F32): arguments are small
  // (|y| < ~450, |log-mag| < ~16) and this kernel is bandwidth-bound.
  const int n   = lane & 15;
  const int mhi = (lane >> 4) << 3;
#pragma unroll
  for (int t = 0; t < 4; ++t) {
    const int col = ohalf * 64 + t * 16 + n;
    const float bv = bias[col];
#pragma unroll
    for (int r = 0; r < 8; ++r) {
      const int row = rowBase + mbase + mhi + r;
      out[(size_t)row * O_DIM + col] = __cosf(accy[t][r] + bv) * __expf(accm[t][r]);
    }
  }
}

extern "C" void kernel_launch(void* const* d_in, const int* in_sizes, int n_in,
                              void* d_out, int out_size, void* d_ws, size_t ws_size,
                              hipStream_t stream) {
  const float* x      = (const float*)d_in[0];
  const float* w_real = (const float*)d_in[1];
  const float* w_imag = (const float*)d_in[2];
  const float* bias   = (const float*)d_in[3];
  float* out = (float*)d_out;
  float* ws  = (float*)d_ws;

  const int B = in_sizes[0] / I_DIM;

  ces_prep_weights<<<(O_DIM * I_DIM + 255) / 256, 256, 0, stream>>>(w_real, w_imag, ws);

  const size_t smem = 2u * O_DIM * I_DIM * sizeof(float); // 128 KB
  ces_main<<<dim3(B / 64), 256, smem, stream>>>(x, ws, bias, out);
}